// SoftMaxlayer_21088289423547
// MI455X (gfx1250) — compile-verified
//
#include <hip/hip_runtime.h>
#include <math.h>

typedef __attribute__((ext_vector_type(2))) float v2f;
typedef __attribute__((ext_vector_type(8))) float v8f;

#define IGNORE_INDEX (-100)
#define BB 64
#define SS 4096
#define LL 128
#define KK 4
#define NTOK (BB * SS)                 // 262144 tokens
#define TOK_PER_WAVE 16
#define WAVES_PER_BLOCK 8
#define TOK_PER_BLOCK (TOK_PER_WAVE * WAVES_PER_BLOCK)  // 128
#define NBLOCKS (NTOK / TOK_PER_BLOCK)                  // 2048

// ---------------------------------------------------------------------------
// Kernel 1: per-sentence first-invalid index (min over s where any target==-100)
// Deterministic: min reduction is order-independent.
// ---------------------------------------------------------------------------
__global__ __launch_bounds__(256) void first_invalid_kernel(
    const int* __restrict__ target, int* __restrict__ firstInv) {
  __shared__ int sm[256];
  const int b = blockIdx.x;
  const int tid = threadIdx.x;
  const int4* tg = (const int4*)target + (size_t)b * SS;  // one int4 per token
  int mn = SS;
  for (int s = tid; s < SS; s += 256) {
    int4 v = tg[s];
    if (v.x == IGNORE_INDEX || v.y == IGNORE_INDEX ||
        v.z == IGNORE_INDEX || v.w == IGNORE_INDEX)
      mn = min(mn, s);
  }
  sm[tid] = mn;
  __syncthreads();
  for (int off = 128; off > 0; off >>= 1) {
    if (tid < off) sm[tid] = min(sm[tid], sm[tid + off]);
    __syncthreads();
  }
  if (tid == 0) firstInv[b] = sm[0];
}

// ---------------------------------------------------------------------------
// Kernel 2: per-token loss via WMMA column sums; one wave32 = 16 tokens.
// Token n = lane&15 is WMMA column n. Lane pair (n, n+16) covers the token's
// 128 scores (2 floats/lane/chunk x 32 chunks). A = ones(16x4) so
// D[m][n] = sum_k B[k][n] + C[m][n]  (exact fp32 K-reduction per column).
// ---------------------------------------------------------------------------
__global__ __launch_bounds__(256) void loss_kernel(
    const float* __restrict__ emit, const int* __restrict__ target,
    const int* __restrict__ firstInv, float* __restrict__ partial) {
  __shared__ float wsum[WAVES_PER_BLOCK];

  const int lane = threadIdx.x & 31;
  const int wave = threadIdx.x >> 5;
  const int n    = lane & 15;   // token within wave group == WMMA column
  const int h    = lane >> 4;   // which half of the K rows this lane feeds
  const int tok  = blockIdx.x * TOK_PER_BLOCK + wave * TOK_PER_WAVE + n;

  // ---- pass 1: load this lane's 64 scores (elements 4c+2h, 4c+2h+1), track max
  const float* row = emit + (size_t)tok * LL + 2 * h;
  v2f v[32];
  float mx = -INFINITY;
#pragma unroll
  for (int c = 0; c < 32; ++c) {
    v[c] = *(const v2f*)(row + 4 * c);
    mx = fmaxf(mx, fmaxf(v[c].x, v[c].y));
  }
  // combine lane pair (n, n+16): both now hold the full per-token max
  mx = fmaxf(mx, __shfl_xor(mx, 16));

  // ---- pass 2: exp and WMMA-accumulate column sums
  v2f ones; ones.x = 1.0f; ones.y = 1.0f;
  v8f acc = {};  // C = 0
#pragma unroll
  for (int c = 0; c < 32; ++c) {
    v2f b;
    b.x = __expf(v[c].x - mx);
    b.y = __expf(v[c].y - mx);
    // D = ones(16x4) x B(4x16) + C : per-column (per-token) exact f32 sums
    acc = __builtin_amdgcn_wmma_f32_16x16x4_f32(
        /*neg_a=*/false, ones, /*neg_b=*/false, b,
        /*c_mod=*/(short)0, acc, /*reuse_a=*/false, /*reuse_b=*/false);
  }
  // All D rows are identical; element 0 holds column (lane&15)'s sum.
  float sum_all = acc[0];

  // ---- gather the K=4 answer scores (2 per lane-half), pair-reduce
  const int* tg = target + (size_t)tok * KK + 2 * h;
  int i0 = tg[0], i1 = tg[1];
  // Clamp: tokens containing IGNORE_INDEX are masked out anyway; avoid OOB/NaN.
  i0 = min(max(i0, 0), LL - 1);
  i1 = min(max(i1, 0), LL - 1);
  const float* rowbase = emit + (size_t)tok * LL;
  float st = __expf(rowbase[i0] - mx) + __expf(rowbase[i1] - mx);
  st += __shfl_xor(st, 16);

  float loss = __logf(sum_all) - __logf(st);

  // ---- prefix-validity mask: token counts iff s < first invalid in sentence
  const int s = tok & (SS - 1);
  const int b_ = tok >> 12;  // S == 4096
  const float valid = (s < firstInv[b_]) ? 1.0f : 0.0f;

  // lanes 16..31 hold duplicate losses -> contribute 0
  float contrib = (h == 0) ? loss * valid : 0.0f;
#pragma unroll
  for (int m = 16; m >= 1; m >>= 1) contrib += __shfl_xor(contrib, m);

  if (lane == 0) wsum[wave] = contrib;
  __syncthreads();
  if (threadIdx.x == 0) {
    float s8 = 0.0f;
    for (int w = 0; w < WAVES_PER_BLOCK; ++w) s8 += wsum[w];  // fixed order
    partial[blockIdx.x] = s8;
  }
}

// ---------------------------------------------------------------------------
// Kernel 3: deterministic final reduction of NBLOCKS partials -> scalar
// ---------------------------------------------------------------------------
__global__ __launch_bounds__(256) void reduce_kernel(
    const float* __restrict__ partial, float* __restrict__ out) {
  __shared__ float sm[256];
  const int tid = threadIdx.x;
  float s = 0.0f;
  for (int i = tid; i < NBLOCKS; i += 256) s += partial[i];  // fixed order
  sm[tid] = s;
  __syncthreads();
  for (int off = 128; off > 0; off >>= 1) {
    if (tid < off) sm[tid] += sm[tid + off];
    __syncthreads();
  }
  if (tid == 0) out[0] = sm[0];
}

// ---------------------------------------------------------------------------
extern "C" void kernel_launch(void* const* d_in, const int* in_sizes, int n_in,
                              void* d_out, int out_size, void* d_ws, size_t ws_size,
                              hipStream_t stream) {
  const float* emit  = (const float*)d_in[0];
  const int*  target = (const int*)d_in[1];

  // ws layout: [0,256) int firstInv[64]; [256, 256+4*NBLOCKS) float partials
  int*   firstInv = (int*)d_ws;
  float* partial  = (float*)((char*)d_ws + 256);

  first_invalid_kernel<<<BB, 256, 0, stream>>>(target, firstInv);
  loss_kernel<<<NBLOCKS, 256, 0, stream>>>(emit, target, firstInv, partial);
  reduce_kernel<<<1, 256, 0, stream>>>(partial, (float*)d_out);
}